// DecodeDetections_20074677141528
// MI455X (gfx1250) — compile-verified
//
#include <hip/hip_runtime.h>

// DecodeDetections: per row of 20 f32 -> 12 f32.
//   out[0:2]  = in[0:2]
//   out[2+2k] = (in[2+2k] * var_w * w + cx) * 512     (k = 0..4)
//   out[3+2k] = (in[3+2k] * var_h * h + cy) * 512
// cx=in[12], cy=in[13], w=in[14], h=in[15], var_w=in[16], var_h=in[17].
//
// Pure streaming kernel: 80 B in + 48 B out per row (both 16B-aligned), so
// all traffic moves as b128 with non-temporal hints. HBM roofline:
// 3.2M rows * 128 B ~= 410 MB @ 23.3 TB/s ~= 17.6 us. No contraction dim and
// the op is nonlinear in the inputs, so WMMA/TDM cannot apply; the win is
// wide NT vmem + global_prefetch_b8 + enough wave32s in flight.

typedef float v4f __attribute__((ext_vector_type(4)));

#define IMG_SCALE 512.0f

__device__ __forceinline__ void decode_row(const float* __restrict__ in,
                                           float* __restrict__ out, int row) {
  // 32-bit index math: row*20 < 2^26, so LLVM can keep SGPR base + 32-bit
  // VGPR offset addressing (no v_mul_u64 in the loop).
  const v4f* rp = (const v4f*)(in + row * 20);
  v4f a = __builtin_nontemporal_load(rp + 0);  // f0  f1  f2  f3
  v4f b = __builtin_nontemporal_load(rp + 1);  // f4  f5  f6  f7
  v4f c = __builtin_nontemporal_load(rp + 2);  // f8  f9  f10 f11
  v4f d = __builtin_nontemporal_load(rp + 3);  // cx  cy  w   h
  v4f e = __builtin_nontemporal_load(rp + 4);  // vw  vh  --  --

  const float cx = d.x;
  const float cy = d.y;
  const float sx = e.x * d.z;  // var_w * w
  const float sy = e.y * d.w;  // var_h * h

  v4f o0, o1, o2;
  o0.x = a.x;
  o0.y = a.y;
  o0.z = (a.z * sx + cx) * IMG_SCALE;
  o0.w = (a.w * sy + cy) * IMG_SCALE;

  o1.x = (b.x * sx + cx) * IMG_SCALE;
  o1.y = (b.y * sy + cy) * IMG_SCALE;
  o1.z = (b.z * sx + cx) * IMG_SCALE;
  o1.w = (b.w * sy + cy) * IMG_SCALE;

  o2.x = (c.x * sx + cx) * IMG_SCALE;
  o2.y = (c.y * sy + cy) * IMG_SCALE;
  o2.z = (c.z * sx + cx) * IMG_SCALE;
  o2.w = (c.w * sy + cy) * IMG_SCALE;

  v4f* op = (v4f*)(out + row * 12);
  __builtin_nontemporal_store(o0, op + 0);
  __builtin_nontemporal_store(o1, op + 1);
  __builtin_nontemporal_store(o2, op + 2);
}

__global__ __launch_bounds__(256) void decode_detections_kernel(
    const float* __restrict__ in, float* __restrict__ out, int n_rows) {
  const int stride = gridDim.x * blockDim.x;
  int row = blockIdx.x * blockDim.x + threadIdx.x;

  // Main loop: two rows per iteration for 2x memory-level parallelism
  // (10 b128 loads in flight per thread).
  for (; row + stride < n_rows; row += 2 * stride) {
    // Prefetch the rows handled two strides ahead (global_prefetch_b8).
    if (row + 2 * stride < n_rows) {
      __builtin_prefetch(in + (row + 2 * stride) * 20, 0, 0);
    }
    if (row + 3 * stride < n_rows) {
      __builtin_prefetch(in + (row + 3 * stride) * 20, 0, 0);
    }
    decode_row(in, out, row);
    decode_row(in, out, row + stride);
  }
  // Tail.
  if (row < n_rows) {
    decode_row(in, out, row);
  }
}

extern "C" void kernel_launch(void* const* d_in, const int* in_sizes, int n_in,
                              void* d_out, int out_size, void* d_ws,
                              size_t ws_size, hipStream_t stream) {
  const float* in = (const float*)d_in[0];
  float* out = (float*)d_out;

  const int n_rows = in_sizes[0] / 20;  // 32 * 100000 = 3.2M rows

  const int block = 256;  // 8 wave32s
  // Cap the grid so each thread iterates a few times (prefetch + unroll have
  // work to do) while keeping >1M threads resident across the device.
  int grid = 4096;
  const int needed = (n_rows + block - 1) / block;
  if (grid > needed) grid = needed;

  decode_detections_kernel<<<grid, block, 0, stream>>>(in, out, n_rows);
}